// get_loss_91276644974603
// MI455X (gfx1250) — compile-verified
//
#include <hip/hip_runtime.h>
#include <hip/hip_bf16.h>
#include <math.h>

#define CONVEX_K  10
#define NUM_CLASS 100
#define B_        8
#define N_        16384
#define M_        (NUM_CLASS * CONVEX_K)   // 1000
#define M_TILES   63                       // ceil(1000/16), 63*16 = 1008
#define N_TILES   (N_ / 16)                // 1024

typedef float v2f __attribute__((ext_vector_type(2)));
typedef float v8f __attribute__((ext_vector_type(8)));

// d_ws layout (floats):
//   [0] = chamfer n-side sum accumulator
//   [1] = convex loss sum accumulator
//   [16 .. 16 + B_*M_) = per-(b,m) min-d2 buffer, viewed as uint bits
#define WS_MINBUF_OFF 16

__global__ void init_ws(float* ws) {
    int i = blockIdx.x * blockDim.x + threadIdx.x;
    if (i < 2) ws[i] = 0.0f;
    unsigned int* mb = (unsigned int*)(ws + WS_MINBUF_OFF);
    if (i < B_ * M_) mb[i] = 0x7F800000u;   // +inf bits (valid min identity for d2 >= 0)
}

__global__ __launch_bounds__(256)
void chamfer_wmma(const float* __restrict__ xyz,
                  const float* __restrict__ skel,
                  float* ws) {
    const int lane = threadIdx.x & 31;
    const int waveId = blockIdx.x * (blockDim.x >> 5) + (threadIdx.x >> 5);
    const int b      = waveId / N_TILES;
    const int n_base = (waveId % N_TILES) * 16;
    const int half   = lane >> 4;    // which 16-lane half
    const int lcol   = lane & 15;

    const float* ptsB = xyz  + (size_t)b * N_ * 6;   // xyz[:, :, :3] of 6 comps
    const float* skB  = skel + (size_t)b * M_ * 3;

    // A operand (16x4 f32): lane holds comps (2*half, 2*half+1) of point n_base+lcol,
    // padded component vector = (x, y, z, 0)
    v2f a;
    {
        const float* p = ptsB + (size_t)(n_base + lcol) * 6;
        a.x = (half == 0) ? p[0] : p[2];
        a.y = (half == 0) ? p[1] : 0.0f;
    }

    // |p|^2 for the 8 rows this lane owns in D: n = n_base + v + 8*half
    float pn2[8];
#pragma unroll
    for (int v = 0; v < 8; ++v) {
        const float* p = ptsB + (size_t)(n_base + v + 8 * half) * 6;
        pn2[v] = p[0]*p[0] + p[1]*p[1] + p[2]*p[2];
    }

    float minm[8];
#pragma unroll
    for (int v = 0; v < 8; ++v) minm[v] = __builtin_inff();

    unsigned int* minbuf = (unsigned int*)(ws + WS_MINBUF_OFF);

    for (int mt = 0; mt < M_TILES; ++mt) {
        const int  m      = mt * 16 + lcol;
        const bool mvalid = (m < M_);

        float sx = 0.0f, sy = 0.0f, sz = 0.0f;
        if (mvalid) {
            const float* s = skB + (size_t)m * 3;
            sx = s[0]; sy = s[1]; sz = s[2];
        }
        const float sm2 = sx*sx + sy*sy + sz*sz;

        // B operand (4x16 f32): lane holds comps (2*half, 2*half+1) of skel point m
        v2f bm;
        bm.x = (half == 0) ? sx : sz;
        bm.y = (half == 0) ? sy : 0.0f;

        v8f c = {0.f, 0.f, 0.f, 0.f, 0.f, 0.f, 0.f, 0.f};
        // D = A(16x4) x B(4x16) + 0 : Gram tile of p.s dot products
        c = __builtin_amdgcn_wmma_f32_16x16x4_f32(false, a, false, bm,
                                                  (short)0, c, false, false);

        float colmin = __builtin_inff();
#pragma unroll
        for (int v = 0; v < 8; ++v) {
            float d2 = fmaf(-2.0f, c[v], pn2[v] + sm2);
            d2 = fmaxf(d2, 1e-12f);
            minm[v] = fminf(minm[v], mvalid ? d2 : __builtin_inff());
            colmin  = fminf(colmin, d2);
        }
        // column m lives in lanes L and L^16; fold halves (min over this wave's 16 n's)
        colmin = fminf(colmin, __shfl_xor(colmin, 16, 32));
        if (lane < 16 && mvalid) {
            union { float f; unsigned int u; } cv; cv.f = colmin;
            atomicMin(&minbuf[(size_t)b * M_ + m], cv.u);   // uint order == float order (>=0)
        }
    }

    // min over m for each n: butterfly over the 16 lanes of each half
#pragma unroll
    for (int v = 0; v < 8; ++v) {
        float mv = minm[v];
        mv = fminf(mv, __shfl_xor(mv, 1, 32));
        mv = fminf(mv, __shfl_xor(mv, 2, 32));
        mv = fminf(mv, __shfl_xor(mv, 4, 32));
        mv = fminf(mv, __shfl_xor(mv, 8, 32));
        minm[v] = mv;
    }
    float s = 0.0f;
#pragma unroll
    for (int v = 0; v < 8; ++v) s += sqrtf(minm[v]);
    if (lcol == 0) atomicAdd(&ws[0], s);   // one add per half (8 n's each)
}

__global__ void convex_loss(const float* __restrict__ skel,
                            const int* __restrict__ labels,
                            float* ws) {
    int t = blockIdx.x * blockDim.x + threadIdx.x;
    if (t >= B_ * NUM_CLASS) return;
    int b = t / NUM_CLASS, cc = t % NUM_CLASS;
    const float* cp  = skel   + ((size_t)b * M_ + cc * CONVEX_K) * 3;
    const int*   lab = labels + (size_t)b * NUM_CLASS * CONVEX_K + cc * CONVEX_K;

    float px[CONVEX_K], py[CONVEX_K], pz[CONVEX_K];
    int   lv[CONVEX_K];
#pragma unroll
    for (int k = 0; k < CONVEX_K; ++k) {
        px[k] = cp[k*3]; py[k] = cp[k*3+1]; pz[k] = cp[k*3+2];
        lv[k] = lab[k];
    }
    float acc = 0.0f;
#pragma unroll
    for (int j = 0; j < CONVEX_K; ++j) {
        float mind = __builtin_inff();
#pragma unroll
        for (int k = 0; k < CONVEX_K; ++k) {
            if (lv[k] == 1) {
                float dx = px[j]-px[k], dy = py[j]-py[k], dz = pz[j]-pz[k];
                mind = fminf(mind, dx*dx + dy*dy + dz*dz);
            }
        }
        mind = fminf(mind, 100000.0f);   // MINDIS_INIT cap
        if (lv[j] != 1) acc += mind;
    }
    atomicAdd(&ws[1], acc);
}

__global__ void finalize_loss(float* ws, float* out) {
    __shared__ float red[256];
    unsigned int* mb = (unsigned int*)(ws + WS_MINBUF_OFF);
    float s = 0.0f;
    for (int i = threadIdx.x; i < B_ * M_; i += 256) {
        union { unsigned int u; float f; } cv; cv.u = mb[i];
        s += sqrtf(cv.f);
    }
    red[threadIdx.x] = s;
    __syncthreads();
    for (int off = 128; off > 0; off >>= 1) {
        if ((int)threadIdx.x < off) red[threadIdx.x] += red[threadIdx.x + off];
        __syncthreads();
    }
    if (threadIdx.x == 0) {
        float loss_convex = ws[1] / (float)(B_ * NUM_CLASS);
        float loss_cd     = (ws[0] + red[0]) / (float)B_;
        out[0] = 1.0f * loss_convex + 0.1f * loss_cd;
    }
}

extern "C" void kernel_launch(void* const* d_in, const int* in_sizes, int n_in,
                              void* d_out, int out_size, void* d_ws, size_t ws_size,
                              hipStream_t stream) {
    const float* xyz    = (const float*)d_in[0];   // [8, 16384, 6] f32
    const float* skel   = (const float*)d_in[1];   // [8, 1000, 3]  f32
    // d_in[2] = weights (unused by reference), d_in[4] = num_class scalar (hardcoded 100)
    const int*   labels = (const int*)d_in[3];     // [8, 100, 10]  i32
    float* ws  = (float*)d_ws;
    float* out = (float*)d_out;

    {   // init accumulators + min buffer
        int n = B_ * M_;                           // 8000 (covers the 2 accumulators too)
        init_ws<<<(n + 255) / 256, 256, 0, stream>>>(ws);
    }
    {   // chamfer: one wave per 16-row n-tile per batch -> 8192 waves, 8 waves/block
        int waves  = B_ * N_TILES;
        int blocks = waves / 8;
        chamfer_wmma<<<blocks, 256, 0, stream>>>(xyz, skel, ws);
    }
    {   // convex hull penalty: one thread per (b, class)
        int n = B_ * NUM_CLASS;
        convex_loss<<<(n + 255) / 256, 256, 0, stream>>>(skel, labels, ws);
    }
    finalize_loss<<<1, 256, 0, stream>>>(ws, out);
}